// TorchSpline_30975304139604
// MI455X (gfx1250) — compile-verified
//
#include <hip/hip_runtime.h>

typedef __attribute__((ext_vector_type(2))) float v2f;
typedef __attribute__((ext_vector_type(8))) float v8f;

#define NKNOT 68   // NC + P + 1 = 64 + 3 + 1

// Cubic Cox-de Boor basis (de Boor "BasisFuns"): returns span k and N[0..3]
// such that N[a] is the basis value for global control index (k-3+a).
// Matches the reference's d[:, p] row exactly (same recurrence, f32).
__device__ __forceinline__ void basis_cubic(const float* __restrict__ t, float x,
                                            int& k, float N[4]) {
    float xc = fminf(fmaxf(x, 0.0f), 1.0f);         // clamp to [t[3], t[64]] = [0,1]
    int i = (int)floorf(xc * 61.0f);                // interior knots are i/61
    i = i < 0 ? 0 : (i > 60 ? 60 : i);
    k = i + 3;                                      // span in [3, 63]
    float left[4], right[4];
    N[0] = 1.0f;
#pragma unroll
    for (int r = 1; r <= 3; ++r) {
        left[r]  = xc - t[k + 1 - r];
        right[r] = t[k + r] - xc;
        float saved = 0.0f;
#pragma unroll
        for (int j = 0; j < r; ++j) {
            float temp = N[j] / (right[j + 1] + left[r - j]);
            N[j] = saved + right[j + 1] * temp;
            saved = left[r - j] * temp;
        }
        N[r] = saved;
    }
}

__device__ __forceinline__ float pick8(v8f v, int g) {
    float r = v[0];
    if (g == 1) r = v[1];
    if (g == 2) r = v[2];
    if (g == 3) r = v[3];
    if (g == 4) r = v[4];
    if (g == 5) r = v[5];
    if (g == 6) r = v[6];
    if (g == 7) r = v[7];
    return r;
}

__global__ void __launch_bounds__(128)
spline3d_wmma_kernel(const float* __restrict__ queries,
                     const float* __restrict__ cp,     // [64*64*64, 3]
                     const float* __restrict__ tx,
                     const float* __restrict__ ty,
                     const float* __restrict__ tz,
                     float* __restrict__ out,          // [NQ, 3]
                     int nq) {
    __shared__ float sk[3 * NKNOT];
    int tid = threadIdx.x;
    if (tid < NKNOT) {
        sk[tid]             = tx[tid];
        sk[NKNOT + tid]     = ty[tid];
        sk[2 * NKNOT + tid] = tz[tid];
    }
    __syncthreads();

    const int lane = tid & 31;
    const int h    = lane >> 4;    // half-wave: K-rows {2h, 2h+1} of each K=4 slice
    const int m    = lane & 15;    // query slot within the wave's 16 queries

    const int wave  = blockIdx.x * 4 + (tid >> 5);   // 4 waves per 128-thread block
    const int qbase = wave * 16;

    // Clamp (not branch) so EXEC stays all-ones for WMMA.
    int q  = qbase + m;
    int qr = q < nq ? q : (nq - 1);

    float xq = queries[qr * 3 + 0];
    float yq = queries[qr * 3 + 1];
    float zq = queries[qr * 3 + 2];

    int kx, ky, kz;
    float bx[4], by[4], bz[4];
    basis_cubic(sk,             xq, kx, bx);
    basis_cubic(sk + NKNOT,     yq, ky, by);
    basis_cubic(sk + 2 * NKNOT, zq, kz, bz);

    const int ix0 = kx - 3 + 2 * h;   // this lane owns x-offsets {2h, 2h+1} (contiguous)
    const int iy0 = ky - 3;
    const int iz0 = kz - 3;
    const float wa0 = bx[2 * h];
    const float wa1 = bx[2 * h + 1];

    v8f accx = {0.f, 0.f, 0.f, 0.f, 0.f, 0.f, 0.f, 0.f};
    v8f accy = {0.f, 0.f, 0.f, 0.f, 0.f, 0.f, 0.f, 0.f};
    v8f accz = {0.f, 0.f, 0.f, 0.f, 0.f, 0.f, 0.f, 0.f};

    // K = 64 contraction as 16 chained K=4 WMMAs; K-slice t = (c,b), inner K = a.
#pragma unroll
    for (int c = 0; c < 4; ++c) {
        const int zoff = (iz0 + c) * 64;
#pragma unroll
        for (int b = 0; b < 4; ++b) {
            const int flat = ix0 + 64 * ((iy0 + b) + zoff);
            const float* p = cp + 3 * flat;           // 2 adjacent points = 6 floats
            const float g0x = p[0], g0y = p[1], g0z = p[2];
            const float g1x = p[3], g1y = p[4], g1z = p[5];

            const float wbc = by[b] * bz[c];
            v2f A;  A.x  = wbc * wa0; A.y  = wbc * wa1;
            v2f Bx; Bx.x = g0x;       Bx.y = g1x;
            v2f By; By.x = g0y;       By.y = g1y;
            v2f Bz; Bz.x = g0z;       Bz.y = g1z;

            accx = __builtin_amdgcn_wmma_f32_16x16x4_f32(false, A, false, Bx,
                                                         (short)0, accx, false, false);
            accy = __builtin_amdgcn_wmma_f32_16x16x4_f32(false, A, false, By,
                                                         (short)0, accy, false, false);
            accz = __builtin_amdgcn_wmma_f32_16x16x4_f32(false, A, false, Bz,
                                                         (short)0, accz, false, false);
        }
    }

    // Diagonal D[m][m] extraction per the 16x16 f32 C/D layout:
    //   m in 0..7  -> lane m      , VGPR m
    //   m in 8..15 -> lane m + 16 , VGPR m - 8
    const bool writer = (lane < 8) || (lane >= 24);
    const int  g      = (lane < 8) ? lane : (lane - 24);
    const int  qm     = (lane < 8) ? lane : (lane - 16);
    const int  qw     = qbase + qm;
    if (writer && qw < nq) {
        out[qw * 3 + 0] = pick8(accx, g);
        out[qw * 3 + 1] = pick8(accy, g);
        out[qw * 3 + 2] = pick8(accz, g);
    }
}

extern "C" void kernel_launch(void* const* d_in, const int* in_sizes, int n_in,
                              void* d_out, int out_size, void* d_ws, size_t ws_size,
                              hipStream_t stream) {
    (void)n_in; (void)out_size; (void)d_ws; (void)ws_size;
    const float* queries = (const float*)d_in[0];
    const float* cp      = (const float*)d_in[1];
    const float* tx      = (const float*)d_in[2];
    const float* ty      = (const float*)d_in[3];
    const float* tz      = (const float*)d_in[4];
    float* out = (float*)d_out;

    const int nq = in_sizes[0] / 3;                 // queries is [NQ, 3]
    const int blocks = (nq + 63) / 64;              // 128 thr = 4 waves = 64 queries
    spline3d_wmma_kernel<<<blocks, 128, 0, stream>>>(queries, cp, tx, ty, tz, out, nq);
}